// SourceModel_5420248727650
// MI455X (gfx1250) — compile-verified
//
#include <hip/hip_runtime.h>
#include <hip/hip_bf16.h>

typedef __attribute__((ext_vector_type(2))) float v2f;
typedef __attribute__((ext_vector_type(8))) float v8f;

#define N_NODES 50000
#define N_EDGES 800000
#define TGT_DD 64
#define EDGE_DD 64
#define UPD_DD 704
#define SLOPE 0.01f

__device__ __forceinline__ float leaky(float x) { return x > 0.f ? x : SLOPE * x; }

__device__ __forceinline__ v8f zero8() {
  v8f z;
#pragma unroll
  for (int i = 0; i < 8; ++i) z[i] = 0.f;
  return z;
}

__device__ __forceinline__ v8f wmma4(v2f a, v2f b, v8f c) {
  // V_WMMA_F32_16X16X4_F32: full-fp32 matrix op
  return __builtin_amdgcn_wmma_f32_16x16x4_f32(false, a, false, b, (short)0, c,
                                               false, false);
}

// --------------------------- CDNA5 async global->LDS -----------------------
// Low 32 bits of a generic __shared__ pointer == wave-relative LDS offset,
// which is what GLOBAL_LOAD_ASYNC_TO_LDS's VDST operand expects.
__device__ __forceinline__ unsigned lds_off(const void* p) {
  return (unsigned)(unsigned long long)p;
}
__device__ __forceinline__ void async_g2l_b128(unsigned ldsoff, const float* g) {
  asm volatile("global_load_async_to_lds_b128 %0, %1, off"
               :: "v"(ldsoff), "v"(g) : "memory");
}
__device__ __forceinline__ void wait_async_0() {
  asm volatile("s_wait_asynccnt 0" ::: "memory");
}
__device__ __forceinline__ void wait_async_2() {
  asm volatile("s_wait_asynccnt 2" ::: "memory");
}
__device__ __forceinline__ void wait_async_3() {
  asm volatile("s_wait_asynccnt 3" ::: "memory");
}

// ---------------------------------------------------------------------------
// Edge MLP inner layer: C(64x128) += A(64x128, LDS stride 132) @ W(128x128).
// W streamed in 16-row chunks through a double-buffered LDS tile (stride 144),
// staged with async global->LDS copies (2 per thread per chunk).
// ---------------------------------------------------------------------------
__device__ __forceinline__ void stage_w_chunk(const float* __restrict__ Wg,
                                              float* lsWbuf, int kc, int tid) {
  const int r = tid >> 4;
  const int c = (tid & 15) * 8;
  const float* g = Wg + (size_t)(kc * 16 + r) * 128 + c;
  const unsigned d = lds_off(&lsWbuf[r * 144 + c]);
  async_g2l_b128(d, g);
  async_g2l_b128(d + 16, g + 4);
}

__device__ __forceinline__ void edge_gemm_128(const float* lsA,
                                              const float* __restrict__ Wg,
                                              float* lsW2,  // [2][16*144]
                                              v8f acc[4], int rt, int cg,
                                              int lane, int tid) {
  const int lrow = lane & 15;
  const int khalf = lane >> 4;
  stage_w_chunk(Wg, lsW2, 0, tid);
#pragma unroll 1
  for (int kc = 0; kc < 8; ++kc) {
    const float* cur = lsW2 + (kc & 1) * (16 * 144);
    if (kc < 7) {
      stage_w_chunk(Wg, lsW2 + ((kc + 1) & 1) * (16 * 144), kc + 1, tid);
      wait_async_2();  // in-order: drains everything except the 2 just issued
    } else {
      wait_async_0();
    }
    __syncthreads();
#pragma unroll
    for (int ks = 0; ks < 4; ++ks) {
      const int k = kc * 16 + ks * 4;
      v2f a = *(const v2f*)&lsA[(rt * 16 + lrow) * 132 + k + khalf * 2];
#pragma unroll
      for (int j = 0; j < 4; ++j) {
        const int n0 = cg * 64 + j * 16;
        v2f b;
        b.x = cur[(ks * 4 + khalf * 2 + 0) * 144 + n0 + lrow];
        b.y = cur[(ks * 4 + khalf * 2 + 1) * 144 + n0 + lrow];
        acc[j] = wmma4(a, b, acc[j]);
      }
    }
    __syncthreads();
  }
}

// ---------------------------------------------------------------------------
// Fused edge-message kernel: gather [x_t[tgt] | edge_attr] -> MLP -> scatter.
// PHASE 0: atomic sum / sumsq / count.  PHASE 1: atomic dev^3 / dev^4.
// 64 edges per block, 256 threads (8 waves).
// ---------------------------------------------------------------------------
template <int PHASE>
__global__ __launch_bounds__(256) void edge_msg_kernel(
    const float* __restrict__ x_t, const int* __restrict__ eidx,
    const float* __restrict__ eattr, const float* __restrict__ mw1,
    const float* __restrict__ mb1, const float* __restrict__ mw2,
    const float* __restrict__ mb2, const float* __restrict__ meanB,
    float* __restrict__ sumB, float* __restrict__ sqB, float* __restrict__ cntB,
    float* __restrict__ d3B, float* __restrict__ d4B) {
  __shared__ float lsIn[64 * 132];
  __shared__ float lsH1[64 * 132];
  __shared__ float lsW2[2 * 16 * 144];
  __shared__ int lsSrc[64];
  __shared__ int lsTgt[64];

  const int tid = threadIdx.x;
  const int eBase = blockIdx.x * 64;
  if (tid < 64) {
    lsSrc[tid] = eidx[eBase + tid];
    lsTgt[tid] = eidx[N_EDGES + eBase + tid];
  }
  __syncthreads();

  {  // async gather of the input tile: cols [0,64)=x_t[tgt], [64,128)=edge_attr
    const int row = tid >> 2;
    const int sub = tid & 3;
    const float* sp;
    if (sub < 2)
      sp = x_t + (size_t)lsTgt[row] * TGT_DD + sub * 32;
    else
      sp = eattr + (size_t)(eBase + row) * EDGE_DD + (sub - 2) * 32;
    const unsigned dp = lds_off(&lsIn[row * 132 + sub * 32]);
#pragma unroll
    for (int i = 0; i < 8; ++i) async_g2l_b128(dp + i * 16, sp + i * 4);
    // drained by the first chunk wait inside edge_gemm_128 (in-order asyncs)
  }

  const int wave = tid >> 5;
  const int lane = tid & 31;
  const int rt = wave & 3;
  const int cg = wave >> 2;
  const int lrow = lane & 15;
  const int khalf = lane >> 4;

  // layer 1
  v8f acc[4];
#pragma unroll
  for (int j = 0; j < 4; ++j) acc[j] = zero8();
  edge_gemm_128(lsIn, mw1, lsW2, acc, rt, cg, lane, tid);

#pragma unroll
  for (int j = 0; j < 4; ++j) {
    const int n = cg * 64 + j * 16 + lrow;
    const float bv = mb1[n];
#pragma unroll
    for (int v = 0; v < 8; ++v) {
      const int m = rt * 16 + v + khalf * 8;
      lsH1[m * 132 + n] = leaky(acc[j][v] + bv);
    }
  }
  __syncthreads();

  // layer 2
#pragma unroll
  for (int j = 0; j < 4; ++j) acc[j] = zero8();
  edge_gemm_128(lsH1, mw2, lsW2, acc, rt, cg, lane, tid);

  // scatter epilogue (L2-resident f32 atomics; accumulators fit in 192MB L2)
#pragma unroll
  for (int j = 0; j < 4; ++j) {
    const int n = cg * 64 + j * 16 + lrow;
    const float bv = mb2[n];
#pragma unroll
    for (int v = 0; v < 8; ++v) {
      const int m = rt * 16 + v + khalf * 8;
      const int s = lsSrc[m];
      const float val = acc[j][v] + bv;
      if (PHASE == 0) {
        atomicAdd(&sumB[(size_t)s * 128 + n], val);
        atomicAdd(&sqB[(size_t)s * 128 + n], val * val);
      } else {
        const float d = val - meanB[(size_t)s * 128 + n];
        const float d2 = d * d;
        atomicAdd(&d3B[(size_t)s * 128 + n], d2 * d);
        atomicAdd(&d4B[(size_t)s * 128 + n], d2 * d2);
      }
    }
  }
  if (PHASE == 0 && tid < 64) atomicAdd(&cntB[lsSrc[tid]], 1.0f);
}

// ---------------------------------------------------------------------------
__global__ __launch_bounds__(256) void zero_kernel(float* __restrict__ p,
                                                   size_t n) {
  size_t i = (size_t)blockIdx.x * 256 + threadIdx.x;
  if (i < n) p[i] = 0.f;
}

__global__ __launch_bounds__(256) void node_stats_kernel(
    const float* __restrict__ sumB, const float* __restrict__ sqB,
    const float* __restrict__ cntB, float* __restrict__ meanB,
    float* __restrict__ stdB) {
  size_t i = (size_t)blockIdx.x * 256 + threadIdx.x;
  if (i >= (size_t)N_NODES * 128) return;
  const int n = (int)(i >> 7);
  const float c = cntB[n];
  float mean = 0.f, msq = 0.f;
  if (c > 0.f) {
    mean = sumB[i] / c;
    msq = sqB[i] / c;
  }
  const float var = leaky(msq - mean * mean);
  meanB[i] = mean;
  stdB[i] = sqrtf(var + 1e-6f);
}

__global__ __launch_bounds__(256) void assemble_kernel(
    const float* __restrict__ x_s, const float* __restrict__ x_u,
    const float* __restrict__ meanB, const float* __restrict__ stdB,
    const float* __restrict__ d3B, const float* __restrict__ d4B,
    const float* __restrict__ cntB, float* __restrict__ h) {
  size_t i = (size_t)blockIdx.x * 256 + threadIdx.x;
  if (i >= (size_t)N_NODES * UPD_DD) return;
  const int n = (int)(i / UPD_DD);
  const int c = (int)(i % UPD_DD);
  float v;
  if (c < 128) {
    v = x_s[(size_t)n * 128 + c];
  } else if (c < 256) {
    v = meanB[(size_t)n * 128 + (c - 128)];
  } else if (c < 384) {
    v = stdB[(size_t)n * 128 + (c - 256)];
  } else if (c < 512) {
    const int cc = c - 384;
    const float cnt = cntB[n];
    const float s = stdB[(size_t)n * 128 + cc];
    v = cnt > 0.f ? (d3B[(size_t)n * 128 + cc] / cnt) / (s * s * s) : 0.f;
  } else if (c < 640) {
    const int cc = c - 512;
    const float cnt = cntB[n];
    const float s = stdB[(size_t)n * 128 + cc];
    const float s2 = s * s;
    v = cnt > 0.f ? (d4B[(size_t)n * 128 + cc] / cnt) / (s2 * s2) : 0.f;
  } else {
    v = x_u[c - 640];
  }
  h[i] = v;
}

// ---------------------------------------------------------------------------
// update GEMM 1: h1 = leaky(h[N,704] @ uw1[704,704] + ub1). Block 128x64.
// Double-buffered async staging: 3 async ops / thread / chunk.
// ---------------------------------------------------------------------------
__device__ __forceinline__ void stage_u1(const float* __restrict__ h,
                                         const float* __restrict__ W,
                                         float* lsA, float* lsB, int mBase,
                                         int nBase, int kc, int tid) {
  {
    const int r = tid >> 1;
    const int half = tid & 1;
    int gr = mBase + r;
    if (gr >= N_NODES) gr = N_NODES - 1;
    const float* g = h + (size_t)gr * UPD_DD + kc * 16 + half * 8;
    const unsigned d = lds_off(&lsA[r * 20 + half * 8]);
    async_g2l_b128(d, g);
    async_g2l_b128(d + 16, g + 4);
  }
  {
    const int r = tid >> 4;
    const int c = (tid & 15) * 4;
    async_g2l_b128(lds_off(&lsB[r * 80 + c]),
                   W + (size_t)(kc * 16 + r) * UPD_DD + nBase + c);
  }
}

__global__ __launch_bounds__(256) void update1_kernel(
    const float* __restrict__ h, const float* __restrict__ W,
    const float* __restrict__ bias, float* __restrict__ h1) {
  __shared__ float lsA[2 * 128 * 20];
  __shared__ float lsB[2 * 16 * 80];
  const int tid = threadIdx.x;
  const int mBase = blockIdx.y * 128;
  const int nBase = blockIdx.x * 64;
  const int wave = tid >> 5;
  const int lane = tid & 31;
  const int lrow = lane & 15;
  const int khalf = lane >> 4;

  v8f acc[4];
#pragma unroll
  for (int j = 0; j < 4; ++j) acc[j] = zero8();

  stage_u1(h, W, lsA, lsB, mBase, nBase, 0, tid);
#pragma unroll 1
  for (int kc = 0; kc < 44; ++kc) {
    const float* curA = lsA + (kc & 1) * (128 * 20);
    const float* curB = lsB + (kc & 1) * (16 * 80);
    if (kc < 43) {
      stage_u1(h, W, lsA + ((kc + 1) & 1) * (128 * 20),
               lsB + ((kc + 1) & 1) * (16 * 80), mBase, nBase, kc + 1, tid);
      wait_async_3();
    } else {
      wait_async_0();
    }
    __syncthreads();
#pragma unroll
    for (int ks = 0; ks < 4; ++ks) {
      v2f a = *(const v2f*)&curA[(wave * 16 + lrow) * 20 + ks * 4 + khalf * 2];
#pragma unroll
      for (int j = 0; j < 4; ++j) {
        v2f b;
        b.x = curB[(ks * 4 + khalf * 2 + 0) * 80 + j * 16 + lrow];
        b.y = curB[(ks * 4 + khalf * 2 + 1) * 80 + j * 16 + lrow];
        acc[j] = wmma4(a, b, acc[j]);
      }
    }
    __syncthreads();
  }
#pragma unroll
  for (int j = 0; j < 4; ++j) {
    const int n = nBase + j * 16 + lrow;
    const float bv = bias[n];
#pragma unroll
    for (int v = 0; v < 8; ++v) {
      const int m = mBase + wave * 16 + v + khalf * 8;
      if (m < N_NODES) h1[(size_t)m * UPD_DD + n] = leaky(acc[j][v] + bv);
    }
  }
}

// ---------------------------------------------------------------------------
// update GEMM 2 + RMSNorm: out = rmsnorm(h1[N,704] @ uw2[704,128] + ub2).
// Block 64x128 so the full output row lives in one block. Async staging.
// ---------------------------------------------------------------------------
__device__ __forceinline__ void stage_u2(const float* __restrict__ h1,
                                         const float* __restrict__ W,
                                         float* lsA, float* lsB, int mBase,
                                         int kc, int tid) {
  {
    const int r = tid >> 2;
    const int q = tid & 3;
    int gr = mBase + r;
    if (gr >= N_NODES) gr = N_NODES - 1;
    async_g2l_b128(lds_off(&lsA[r * 20 + q * 4]),
                   h1 + (size_t)gr * UPD_DD + kc * 16 + q * 4);
  }
  {
    const int r = tid >> 4;
    const int c = (tid & 15) * 8;
    const float* g = W + (size_t)(kc * 16 + r) * 128 + c;
    const unsigned d = lds_off(&lsB[r * 144 + c]);
    async_g2l_b128(d, g);
    async_g2l_b128(d + 16, g + 4);
  }
}

__global__ __launch_bounds__(256) void update2_kernel(
    const float* __restrict__ h1, const float* __restrict__ W,
    const float* __restrict__ bias, const float* __restrict__ norm_w,
    float* __restrict__ out) {
  __shared__ float lsA[2 * 64 * 20];
  __shared__ float lsB[2 * 16 * 144];
  __shared__ float lsP[64 * 132];
  __shared__ float lsR[64];
  const int tid = threadIdx.x;
  const int mBase = blockIdx.x * 64;
  const int wave = tid >> 5;
  const int lane = tid & 31;
  const int rt = wave & 3;
  const int cg = wave >> 2;
  const int lrow = lane & 15;
  const int khalf = lane >> 4;

  v8f acc[4];
#pragma unroll
  for (int j = 0; j < 4; ++j) acc[j] = zero8();

  stage_u2(h1, W, lsA, lsB, mBase, 0, tid);
#pragma unroll 1
  for (int kc = 0; kc < 44; ++kc) {
    const float* curA = lsA + (kc & 1) * (64 * 20);
    const float* curB = lsB + (kc & 1) * (16 * 144);
    if (kc < 43) {
      stage_u2(h1, W, lsA + ((kc + 1) & 1) * (64 * 20),
               lsB + ((kc + 1) & 1) * (16 * 144), mBase, kc + 1, tid);
      wait_async_3();
    } else {
      wait_async_0();
    }
    __syncthreads();
#pragma unroll
    for (int ks = 0; ks < 4; ++ks) {
      v2f a = *(const v2f*)&curA[(rt * 16 + lrow) * 20 + ks * 4 + khalf * 2];
#pragma unroll
      for (int j = 0; j < 4; ++j) {
        const int n0 = cg * 64 + j * 16;
        v2f b;
        b.x = curB[(ks * 4 + khalf * 2 + 0) * 144 + n0 + lrow];
        b.y = curB[(ks * 4 + khalf * 2 + 1) * 144 + n0 + lrow];
        acc[j] = wmma4(a, b, acc[j]);
      }
    }
    __syncthreads();
  }

#pragma unroll
  for (int j = 0; j < 4; ++j) {
    const int n = cg * 64 + j * 16 + lrow;
    const float bv = bias[n];
#pragma unroll
    for (int v = 0; v < 8; ++v) {
      const int m = rt * 16 + v + khalf * 8;
      lsP[m * 132 + n] = acc[j][v] + bv;
    }
  }
  __syncthreads();

  if (tid < 64) {
    float ss = 0.f;
#pragma unroll
    for (int c = 0; c < 128; ++c) {
      const float x = lsP[tid * 132 + c];
      ss += x * x;
    }
    lsR[tid] = rsqrtf(ss * (1.f / 128.f) + 1.1920928955078125e-7f);
  }
  __syncthreads();

  for (int i = tid; i < 64 * 128; i += 256) {
    const int r = i >> 7;
    const int c = i & 127;
    const int m = mBase + r;
    if (m < N_NODES)
      out[(size_t)m * 128 + c] = lsP[r * 132 + c] * lsR[r] * norm_w[c];
  }
}

// ---------------------------------------------------------------------------
extern "C" void kernel_launch(void* const* d_in, const int* in_sizes, int n_in,
                              void* d_out, int out_size, void* d_ws,
                              size_t ws_size, hipStream_t stream) {
  const float* x_s = (const float*)d_in[0];
  const float* x_t = (const float*)d_in[1];
  const int* eidx = (const int*)d_in[2];
  const float* eattr = (const float*)d_in[3];
  const float* x_u = (const float*)d_in[4];
  const float* mw1 = (const float*)d_in[5];
  const float* mb1 = (const float*)d_in[6];
  const float* mw2 = (const float*)d_in[7];
  const float* mb2 = (const float*)d_in[8];
  const float* uw1 = (const float*)d_in[9];
  const float* ub1 = (const float*)d_in[10];
  const float* uw2 = (const float*)d_in[11];
  const float* ub2 = (const float*)d_in[12];
  const float* norm_w = (const float*)d_in[13];
  float* out = (float*)d_out;

  const size_t NF = (size_t)N_NODES * 128;
  float* ws = (float*)d_ws;
  float* sumB = ws;
  float* sqB = ws + NF;
  float* d3B = ws + 2 * NF;
  float* d4B = ws + 3 * NF;
  float* cntB = ws + 4 * NF;                    // N floats
  float* meanB = ws + 4 * NF + N_NODES;
  float* stdB = meanB + NF;
  float* hB = stdB + NF;                        // N*704
  float* h1B = hB + (size_t)N_NODES * UPD_DD;   // N*704

  // 1. zero the atomic accumulators (sum, sq, dev3, dev4, count)
  const size_t nz = 4 * NF + N_NODES;
  zero_kernel<<<(unsigned)((nz + 255) / 256), 256, 0, stream>>>(ws, nz);

  // 2. edge MLP + scatter sum/sumsq/count
  edge_msg_kernel<0><<<N_EDGES / 64, 256, 0, stream>>>(
      x_t, eidx, eattr, mw1, mb1, mw2, mb2, meanB, sumB, sqB, cntB, d3B, d4B);

  // 3. mean / std
  node_stats_kernel<<<(unsigned)((NF + 255) / 256), 256, 0, stream>>>(
      sumB, sqB, cntB, meanB, stdB);

  // 4. recompute edge MLP + scatter dev^3 / dev^4
  edge_msg_kernel<1><<<N_EDGES / 64, 256, 0, stream>>>(
      x_t, eidx, eattr, mw1, mb1, mw2, mb2, meanB, sumB, sqB, cntB, d3B, d4B);

  // 5. assemble h = [x_s | mean | std | skew | kurt | x_u]
  const size_t nh = (size_t)N_NODES * UPD_DD;
  assemble_kernel<<<(unsigned)((nh + 255) / 256), 256, 0, stream>>>(
      x_s, x_u, meanB, stdB, d3B, d4B, cntB, hB);

  // 6. h1 = leaky(h @ uw1 + ub1)   [50000 x 704]
  dim3 g1(UPD_DD / 64, (N_NODES + 127) / 128);
  update1_kernel<<<g1, 256, 0, stream>>>(hB, uw1, ub1, h1B);

  // 7. out = rmsnorm(h1 @ uw2 + ub2) * norm_w   [50000 x 128]
  update2_kernel<<<(N_NODES + 63) / 64, 256, 0, stream>>>(h1B, uw2, ub2,
                                                          norm_w, out);
}